// HIMALAYAAdapter_56538949484761
// MI455X (gfx1250) — compile-verified
//
#include <hip/hip_runtime.h>
#include <math.h>

// Problem constants (from reference): B=32, T=2048, H=1024, WIDTH=32, TOTAL=512, top-k=8
#define B_   32
#define T_   2048
#define H_   1024
#define WID_ 32
#define TOT_ 512
#define KTOP 8

typedef __attribute__((ext_vector_type(2))) float v2f;
typedef __attribute__((ext_vector_type(4))) float v4f;
typedef __attribute__((ext_vector_type(8))) float v8f;

// ---------------------------------------------------------------------------
// Kernel 1: fused head. One workgroup (8 wave32s).
//   stage 1: h1 = relu(cls @ W1 + b1)            32x32,  K=1024  (WMMA f32)
//   stage 2: logits = (h1 @ W2 + b2)/|temp|      32x512, K=32    (WMMA f32)
//   stage 3: per-row max/sum-exp + top-8 (softmax is monotone -> top-k on logits)
//   stage 4: delta[b,h] = normalize(sum_j coeff_j * dict[idx_j, h]) / sqrt(H)
// ---------------------------------------------------------------------------
__global__ __launch_bounds__(256) void head_kernel(
    const float* __restrict__ hidden, const float* __restrict__ W1,
    const float* __restrict__ b1,     const float* __restrict__ W2,
    const float* __restrict__ b2,     const float* __restrict__ D_c,
    const float* __restrict__ D_e,    const float* __restrict__ temp,
    float* __restrict__ delta)
{
  __shared__ float s_h1[B_][WID_];        //  4 KB
  __shared__ float s_logits[B_][TOT_];    // 64 KB
  __shared__ float s_topv[B_ * KTOP];     //  1 KB
  __shared__ int   s_topi[B_ * KTOP];     //  1 KB
  __shared__ float s_rowsq[B_];

  const int tid  = threadIdx.x;
  const int lane = tid & 31;
  const int wave = tid >> 5;
  const int nloc = lane & 15;   // N (or M for A-matrix) position within 16
  const int hi   = lane >> 4;   // half-wave select (K split for A/B tiles)

  if (tid < B_) s_rowsq[tid] = 0.0f;

  // ---- Stage 1: h1 = relu(cls @ W1 + b1). 2x2 tiles of 16x16, waves 0..3.
  if (wave < 4) {
    const int mi = (wave & 1) * 16;
    const int ni = (wave >> 1) * 16;
    const int m  = mi + nloc;          // A row this lane supplies
    const int n  = ni + nloc;          // B col this lane supplies
    v8f c = {};
    for (int k = 0; k < H_; k += 4) {
      // A 16x4 f32: v[0] = K = 2*hi, v[1] = K = 2*hi+1 (cls row m, contiguous)
      const float* ap = hidden + (size_t)m * (T_ * H_) + (k + 2 * hi);
      v2f a; a.x = ap[0]; a.y = ap[1];
      // B 4x16 f32: v[0] = row K = 2*hi, v[1] = row K = 2*hi+1, col n
      const float* bp = W1 + (size_t)(k + 2 * hi) * WID_ + n;
      v2f b; b.x = bp[0]; b.y = bp[WID_];
      c = __builtin_amdgcn_wmma_f32_16x16x4_f32(false, a, false, b,
                                                (short)0, c, false, false);
    }
    #pragma unroll
    for (int r = 0; r < 8; ++r) {      // D: VGPR r holds M=r (lanes 0-15), M=r+8 (16-31)
      const int mm = mi + r + 8 * hi;
      const float v = c[r] + b1[n];
      s_h1[mm][n] = v > 0.0f ? v : 0.0f;
    }
  }
  __syncthreads();

  // ---- Stage 2: logits = (h1 @ W2 + b2) / |temp|. 64 tiles, 8 per wave.
  const float invT = 1.0f / fabsf(temp[0]);
  for (int j = 0; j < 8; ++j) {
    const int tile = wave * 8 + j;
    const int mi = (tile & 1) * 16;
    const int ni = (tile >> 1) * 16;
    const int m  = mi + nloc;
    const int n  = ni + nloc;
    v8f c = {};
    #pragma unroll
    for (int k = 0; k < WID_; k += 4) {
      v2f a; a.x = s_h1[m][k + 2 * hi]; a.y = s_h1[m][k + 2 * hi + 1];
      const float* bp = W2 + (size_t)(k + 2 * hi) * TOT_ + n;
      v2f b; b.x = bp[0]; b.y = bp[TOT_];
      c = __builtin_amdgcn_wmma_f32_16x16x4_f32(false, a, false, b,
                                                (short)0, c, false, false);
    }
    #pragma unroll
    for (int r = 0; r < 8; ++r) {
      const int mm = mi + r + 8 * hi;
      s_logits[mm][n] = (c[r] + b2[n]) * invT;
    }
  }
  __syncthreads();

  // ---- Stage 3: softmax stats + top-8 per row (4 rows per wave).
  for (int rr = 0; rr < 4; ++rr) {
    const int row = wave * 4 + rr;
    float vals[16];
    #pragma unroll
    for (int q = 0; q < 16; ++q) vals[q] = s_logits[row][lane + 32 * q];

    float rmax = -INFINITY;
    #pragma unroll
    for (int q = 0; q < 16; ++q) rmax = fmaxf(rmax, vals[q]);
    for (int off = 16; off; off >>= 1)
      rmax = fmaxf(rmax, __shfl_xor(rmax, off, 32));

    float rsum = 0.0f;
    #pragma unroll
    for (int q = 0; q < 16; ++q) rsum += expf(vals[q] - rmax);
    for (int off = 16; off; off >>= 1)
      rsum += __shfl_xor(rsum, off, 32);

    for (int it = 0; it < KTOP; ++it) {
      float bv = -INFINITY; int bi = TOT_;
      #pragma unroll
      for (int q = 0; q < 16; ++q) {
        const int nn = lane + 32 * q;
        if (vals[q] > bv) { bv = vals[q]; bi = nn; }
      }
      for (int off = 16; off; off >>= 1) {
        const float ov = __shfl_xor(bv, off, 32);
        const int   oi = __shfl_xor(bi, off, 32);
        if (ov > bv || (ov == bv && oi < bi)) { bv = ov; bi = oi; }
      }
      if (lane == 0) {
        s_topv[row * KTOP + it] = expf(bv - rmax) / rsum;  // = prob of winner
        s_topi[row * KTOP + it] = bi;
      }
      if ((bi & 31) == lane) vals[bi >> 5] = -INFINITY;    // knock out winner
    }
  }
  __syncthreads();

  // ---- Stage 4: delta = normalize(coeff @ dict) / 32. 8 threads per row, 128 cols each.
  {
    const int r  = tid >> 3;
    const int h0 = (tid & 7) * (H_ / 8);
    float cv[KTOP]; const float* dp[KTOP];
    #pragma unroll
    for (int j = 0; j < KTOP; ++j) {
      cv[j] = s_topv[r * KTOP + j];
      const int ix = s_topi[r * KTOP + j];
      dp[j] = (ix < 256) ? (D_c + (size_t)ix * H_) : (D_e + (size_t)(ix - 256) * H_);
    }
    float sumsq = 0.0f;
    for (int h = 0; h < H_ / 8; h += 4) {
      v4f u = {};
      #pragma unroll
      for (int j = 0; j < KTOP; ++j) {
        const v4f dv = *(const v4f*)(dp[j] + h0 + h);
        u += cv[j] * dv;
      }
      *(v4f*)(delta + (size_t)r * H_ + h0 + h) = u;
      sumsq += u.x * u.x + u.y * u.y + u.z * u.z + u.w * u.w;
    }
    atomicAdd(&s_rowsq[r], sumsq);
    __syncthreads();
    const float inv = (1.0f / (sqrtf(s_rowsq[r]) + 1e-12f)) * 0.03125f; // /||u|| /sqrt(H)
    for (int h = 0; h < H_ / 8; h += 4) {
      v4f u = *(v4f*)(delta + (size_t)r * H_ + h0 + h);
      u *= inv;
      *(v4f*)(delta + (size_t)r * H_ + h0 + h) = u;
    }
  }
}

// ---------------------------------------------------------------------------
// Kernel 2: out[b,t,:] = hidden[b,t,:] + delta[b,:].  512 MB stream at HBM rate.
// b128 loads/stores, non-temporal (single-use stream; avoid wasting 192MB L2).
// Block = 256 threads covers one full H row per iteration; 32 rows per block.
// Grid = B * (T/32) = 2048 blocks.
// ---------------------------------------------------------------------------
__global__ __launch_bounds__(256) void bcast_add_kernel(
    const float* __restrict__ hidden, const float* __restrict__ delta,
    float* __restrict__ out)
{
  const int b     = blockIdx.x >> 6;        // T/32 = 64 chunks per batch row
  const int chunk = blockIdx.x & 63;
  const size_t base = (((size_t)b * T_) + (size_t)chunk * 32) * H_;

  const v4f d = *((const v4f*)(delta + (size_t)b * H_) + threadIdx.x);
  const v4f* __restrict__ src = (const v4f*)(hidden + base) + threadIdx.x;
  v4f* __restrict__ dst       = (v4f*)(out + base) + threadIdx.x;

  #pragma unroll 4
  for (int t = 0; t < 32; ++t) {
    v4f x = __builtin_nontemporal_load(src);
    x += d;
    __builtin_nontemporal_store(x, dst);
    src += H_ / 4;
    dst += H_ / 4;
  }
}

extern "C" void kernel_launch(void* const* d_in, const int* in_sizes, int n_in,
                              void* d_out, int out_size, void* d_ws, size_t ws_size,
                              hipStream_t stream) {
  const float* hidden = (const float*)d_in[0];
  const float* W1     = (const float*)d_in[1];
  const float* b1     = (const float*)d_in[2];
  const float* W2     = (const float*)d_in[3];
  const float* b2     = (const float*)d_in[4];
  const float* D_c    = (const float*)d_in[5];
  const float* D_e    = (const float*)d_in[6];
  const float* temp   = (const float*)d_in[7];
  float* out   = (float*)d_out;
  float* delta = (float*)d_ws;  // 32*1024 floats = 128 KB scratch

  head_kernel<<<1, 256, 0, stream>>>(hidden, W1, b1, W2, b2, D_c, D_e, temp, delta);
  bcast_add_kernel<<<B_ * (T_ / 32), 256, 0, stream>>>(hidden, delta, out);
}